// TrilinearUpsample3d_16673063043678
// MI455X (gfx1250) — compile-verified
//
#include <hip/hip_runtime.h>

#define GRID_DIM 128

typedef __attribute__((ext_vector_type(2))) float v2f;
typedef __attribute__((ext_vector_type(8))) float v8f;

// Global-address-space pointers: keep the gathered feature-row loads on the
// global_load path (LOADcnt only) instead of flat_load (LOADcnt + DScnt).
typedef const __attribute__((address_space(1))) float* gcfp;
typedef __attribute__((address_space(1))) float* gfp;

// ---------------- hash table kernels (replaces argsort+searchsorted) --------

__global__ void ht_clear_kernel(int* __restrict__ htK, int tab) {
  int t = blockIdx.x * blockDim.x + threadIdx.x;
  if (t < tab) htK[t] = -1;
}

__global__ void ht_insert_kernel(const int* __restrict__ coords,
                                 int* __restrict__ htK, int* __restrict__ htV,
                                 int n, unsigned hshift, unsigned hmask) {
  int i = blockIdx.x * blockDim.x + threadIdx.x;
  if (i >= n) return;
  int b = coords[i * 4 + 0], x = coords[i * 4 + 1];
  int y = coords[i * 4 + 2], z = coords[i * 4 + 3];
  int key = ((b * GRID_DIM + x) * GRID_DIM + y) * GRID_DIM + z;
  unsigned h = ((unsigned)key * 0x9E3779B1u) >> hshift;
  for (;;) {
    int prev = atomicCAS(&htK[h], -1, key);
    if (prev == -1) { htV[h] = i; break; }
    h = (h + 1) & hmask;
  }
}

// ---------------- main kernel: out[16x64] = A[16x56] x G[56x64] via WMMA ----
// Wave handles 2 voxels. K layout: k in [0,28) -> voxel0 corner k (27 = pad),
// k in [28,56) -> voxel1 corner k-28. A is the constant trilinear weight
// matrix (block diagonal over the 2 voxels). B rows are gathered feature rows
// (or the default-feature row when absent / out of bounds / pad).

__launch_bounds__(128)
__global__ void upsample_kernel(const int* __restrict__ coords,
                                const float* __restrict__ feats,
                                const float* __restrict__ defF,
                                const int* __restrict__ htK,
                                const int* __restrict__ htV,
                                float* __restrict__ out,
                                int n, unsigned hshift, unsigned hmask) {
  __shared__ float sA[28 * 32];                 // [(k>>1)][m][k&1] weight pairs
  __shared__ unsigned long long sRow[4 * 56];   // per-wave row pointers

  const int tid  = threadIdx.x;
  const int lane = tid & 31;
  const int wid  = __builtin_amdgcn_readfirstlane(tid >> 5);

  // Build constant A matrix in LDS (once per block).
  for (int i = tid; i < 28 * 32; i += 128) {
    int k  = ((i >> 5) << 1) | (i & 1);
    int m  = (i >> 1) & 15;
    int vk = (k >= 28) ? 1 : 0;
    int c  = k - vk * 28;
    int vm = m >> 3;
    int o  = m & 7;
    float w = 0.0f;
    if (vm == vk && c < 27) {
      int cx = c / 9, cy = (c / 3) % 3, cz = c % 3;
      int ox = (o >> 2) & 1, oy = (o >> 1) & 1, oz = o & 1;
      float wx = (cx == 1) ? 0.75f : ((cx == ox * 2) ? 0.25f : 0.0f);
      float wy = (cy == 1) ? 0.75f : ((cy == oy * 2) ? 0.25f : 0.0f);
      float wz = (cz == 1) ? 0.75f : ((cz == oz * 2) ? 0.25f : 0.0f);
      w = wx * wy * wz;
    }
    sA[i] = w;
  }

  // Phase A: hash-lookup the 27-neighborhood of each of this wave's 2 voxels.
  const long long i0 = (long long)blockIdx.x * 8 + wid * 2;
  for (int v = 0; v < 2; ++v) {
    long long vi = i0 + v;
    if (lane < 28) {
      unsigned long long p = (unsigned long long)(const void*)defF;
      if (lane < 27 && vi < n) {
        int b = coords[vi * 4 + 0], x = coords[vi * 4 + 1];
        int y = coords[vi * 4 + 2], z = coords[vi * 4 + 3];
        int c  = lane;
        int nx = x - 1 + c / 9;
        int ny = y - 1 + (c / 3) % 3;
        int nz = z - 1 + c % 3;
        if ((unsigned)nx < (unsigned)GRID_DIM &&
            (unsigned)ny < (unsigned)GRID_DIM &&
            (unsigned)nz < (unsigned)GRID_DIM) {
          int key = ((b * GRID_DIM + nx) * GRID_DIM + ny) * GRID_DIM + nz;
          unsigned h = ((unsigned)key * 0x9E3779B1u) >> hshift;
          int idx = -1;
          for (;;) {
            int k2 = htK[h];
            if (k2 == key) { idx = htV[h]; break; }
            if (k2 == -1) break;
            h = (h + 1) & hmask;
          }
          if (idx >= 0)
            p = (unsigned long long)(const void*)(feats + (long long)idx * 64);
        }
      }
      sRow[wid * 56 + v * 28 + lane] = p;
    }
  }
  __syncthreads();

  // Phase B: 14 K-steps x 4 column tiles of V_WMMA_F32_16X16X4_F32.
  // A/B lane layout (f32 16x16x4): vgpr j, lane L: k_local = j + 2*(L/16),
  // m (A) / n (B) = L%16.
  const int half = (lane >> 4) & 1;
  const int m    = lane & 15;

  v8f acc0 = {}; v8f acc1 = {}; v8f acc2 = {}; v8f acc3 = {};
  const unsigned long long* rp = &sRow[wid * 56];

#pragma unroll
  for (int t = 0; t < 14; ++t) {
    const int k0 = 4 * t + 2 * half;
    v2f Av = *(const v2f*)&sA[(2 * t + half) * 32 + 2 * m];
    unsigned long long p0 = rp[k0];
    unsigned long long p1 = rp[k0 + 1];
    gcfp q0 = (gcfp)p0 + m;   // addrspace(1): lower as global_load_b32
    gcfp q1 = (gcfp)p1 + m;
    v2f B0 = { q0[0],  q1[0]  };
    v2f B1 = { q0[16], q1[16] };
    v2f B2 = { q0[32], q1[32] };
    v2f B3 = { q0[48], q1[48] };
    acc0 = __builtin_amdgcn_wmma_f32_16x16x4_f32(false, Av, false, B0, (short)0, acc0, false, false);
    acc1 = __builtin_amdgcn_wmma_f32_16x16x4_f32(false, Av, false, B1, (short)0, acc1, false, false);
    acc2 = __builtin_amdgcn_wmma_f32_16x16x4_f32(false, Av, false, B2, (short)0, acc2, false, false);
    acc3 = __builtin_amdgcn_wmma_f32_16x16x4_f32(false, Av, false, B3, (short)0, acc3, false, false);
  }

  // Store: D vgpr r -> lanes 0-15 row r (voxel0), lanes 16-31 row r+8 (voxel1).
  if (i0 + half < n) {
    gfp ob = (gfp)(out + (i0 * 8 + (long long)half * 8) * 64 + m);
#pragma unroll
    for (int r = 0; r < 8; ++r) {
      ob[r * 64 +  0] = acc0[r];
      ob[r * 64 + 16] = acc1[r];
      ob[r * 64 + 32] = acc2[r];
      ob[r * 64 + 48] = acc3[r];
    }
  }
}

// ---------------- launch ----------------------------------------------------

extern "C" void kernel_launch(void* const* d_in, const int* in_sizes, int n_in,
                              void* d_out, int out_size, void* d_ws, size_t ws_size,
                              hipStream_t stream) {
  const int*   coords = (const int*)d_in[0];
  const float* feats  = (const float*)d_in[1];
  const float* defF   = (const float*)d_in[2];
  float* out = (float*)d_out;
  const int n = in_sizes[0] / 4;   // coords is (N,4) int32

  // Hash table sized to workspace (power of two, >= 2^17 entries).
  int logTab = 19;
  while (logTab > 17 && (((size_t)8) << logTab) > ws_size) --logTab;
  const int tab = 1 << logTab;
  const unsigned hshift = (unsigned)(32 - logTab);
  const unsigned hmask  = (unsigned)(tab - 1);
  int* htK = (int*)d_ws;
  int* htV = htK + tab;

  ht_clear_kernel<<<(tab + 255) / 256, 256, 0, stream>>>(htK, tab);
  ht_insert_kernel<<<(n + 255) / 256, 256, 0, stream>>>(coords, htK, htV, n,
                                                        hshift, hmask);
  const int nblocks = (n + 7) / 8;   // 8 voxels per 128-thread block
  upsample_kernel<<<nblocks, 128, 0, stream>>>(coords, feats, defF, htK, htV,
                                               out, n, hshift, hmask);
}